// Net_Actor_44890998178489
// MI455X (gfx1250) — compile-verified
//
#include <hip/hip_runtime.h>
#include <hip/hip_bf16.h>
#include <math.h>

// ---------------------------------------------------------------------------
// Sizes (match reference)
// ---------------------------------------------------------------------------
#define NPG   1024            // nodes per graph
#define BATCH 32
#define NTOT  (NPG * BATCH)   // 32768
#define D     128
#define NEG_SLOPE 0.2f

typedef float v2f __attribute__((ext_vector_type(2)));
typedef float v8f __attribute__((ext_vector_type(8)));

// ---------------------------------------------------------------------------
// WMMA f32 GEMM:  C[M,128] = A[M,128] @ W[128,128] (+ bias)
// Block: 256 threads = 8 waves; each wave owns a 16-row strip, 8 n-tiles.
//
// W is staged in LDS in k-pair-interleaved layout:
//   lw[(k>>1)*PSTRF + col*2 + (k&1)] = W[k][col]
// so a lane's B fragment (W[k][col], W[k+1][col]) is one aligned 8-byte
// ds_load_b64 straight into an even VGPR pair (no register shuffling).
// PSTRF = 288 floats: the +32 pad puts lanes 16-31 (k+2 pair) on the
// opposite 32-bank half -> conflict-free b64 reads.
// ---------------------------------------------------------------------------
#define PSTRF 288   // floats per k-pair row (128 pairs * 2 + 32 pad)

__global__ __launch_bounds__(256)
void gemm128_wmma(const float* __restrict__ A, const float* __restrict__ W,
                  const float* __restrict__ bias, float* __restrict__ C, int M)
{
    __shared__ float lw[64 * PSTRF];   // 73728 B
    const int t = threadIdx.x;

    // Cooperative load of W (16384 floats) into pair-interleaved LDS.
    for (int i = t; i < 128 * 32; i += 256) {
        const int row = i >> 5;           // k   (0..127)
        const int c4  = (i & 31) << 2;    // col (multiple of 4)
        const float4 wv = *reinterpret_cast<const float4*>(W + row * 128 + c4);
        float* dst = &lw[(row >> 1) * PSTRF + c4 * 2 + (row & 1)];
        dst[0] = wv.x; dst[2] = wv.y; dst[4] = wv.z; dst[6] = wv.w;
    }
    __syncthreads();

    const int wave = t >> 5;
    const int lane = t & 31;
    const int m0   = blockIdx.x * 128 + wave * 16;
    if (m0 >= M) return;

    const int half = lane >> 4;     // 0: lanes 0-15, 1: lanes 16-31
    const int l15  = lane & 15;

    v8f acc[8];
    for (int n = 0; n < 8; ++n) {
        const float b = bias ? bias[n * 16 + l15] : 0.0f;
        acc[n] = (v8f){b, b, b, b, b, b, b, b};
    }

    // A fragment (16x4 f32 = 2 VGPRs): lane(l15,half) supplies
    // A[m0+l15][kk*4 + half*2 + {0,1}]  -> one global b64 per k-step.
    const float* arow = A + (size_t)(m0 + l15) * 128 + half * 2;

    for (int kk = 0; kk < 32; ++kk) {
        const v2f a = *reinterpret_cast<const v2f*>(arow + kk * 4);
        // B fragment base: k-pair index (kk*2 + half) holds rows
        // {kk*4+half*2, kk*4+half*2+1}; column n*16 + l15.
        const float* wp = &lw[(kk * 2 + half) * PSTRF + l15 * 2];
        #pragma unroll
        for (int n = 0; n < 8; ++n) {
            const v2f b = *reinterpret_cast<const v2f*>(wp + n * 32);
            acc[n] = __builtin_amdgcn_wmma_f32_16x16x4_f32(
                false, a, false, b, (short)0, acc[n], false, false);
        }
    }

    // Store: acc[n][r] -> row m0 + half*8 + r, col n*16 + l15.
    float* crow = C + (size_t)(m0 + half * 8) * 128 + l15;
    #pragma unroll
    for (int n = 0; n < 8; ++n)
        #pragma unroll
        for (int r = 0; r < 8; ++r)
            crow[(size_t)r * 128 + n * 16] = acc[n][r];
}

// ---------------------------------------------------------------------------
// Per-layer init: seg-max = -inf, seg-sum = 0, output accumulator = 0
// ---------------------------------------------------------------------------
__global__ __launch_bounds__(256)
void init_layer(float* __restrict__ m, float* __restrict__ s,
                float* __restrict__ o, size_t n)
{
    size_t i = (size_t)blockIdx.x * 256 + threadIdx.x;
    if (i < n) { m[i] = -INFINITY; s[i] = 0.0f; o[i] = 0.0f; }
}

// ---------------------------------------------------------------------------
// Float atomic max via sign-split int punning (native int atomics)
// ---------------------------------------------------------------------------
__device__ __forceinline__ void atomicMaxF32(float* addr, float v)
{
    if (v >= 0.0f) atomicMax((int*)addr, __float_as_int(v));
    else           atomicMin((unsigned int*)addr, __float_as_uint(v));
}

__device__ __forceinline__ float lrelu(float x)
{
    return x > 0.0f ? x : x * NEG_SLOPE;
}

// One wave per edge, one float4 of channels per lane (fully coalesced 512B).
__device__ __forceinline__ void edge_decode(int e, int E,
                                            const int* __restrict__ esrc,
                                            const int* __restrict__ edst,
                                            int& s, int& d)
{
    if (e < E) { s = esrc[e]; d = edst[e]; }
    else       { s = d = e - E; }          // appended self loops
}

__global__ __launch_bounds__(256)
void edge_segmax(const int* __restrict__ esrc, const int* __restrict__ edst,
                 const float* __restrict__ p, const float* __restrict__ q,
                 float* __restrict__ m, int E, int NE)
{
    const int e = blockIdx.x * 8 + (threadIdx.x >> 5);
    if (e >= NE) return;
    int s, d; edge_decode(e, E, esrc, edst, s, d);
    const int c = (threadIdx.x & 31) * 4;
    const float4 pv = *reinterpret_cast<const float4*>(p + (size_t)d * D + c);
    const float4 qv = *reinterpret_cast<const float4*>(q + (size_t)s * D + c);
    float* mb = m + (size_t)d * D + c;
    atomicMaxF32(mb + 0, lrelu(pv.x + qv.x));
    atomicMaxF32(mb + 1, lrelu(pv.y + qv.y));
    atomicMaxF32(mb + 2, lrelu(pv.z + qv.z));
    atomicMaxF32(mb + 3, lrelu(pv.w + qv.w));
}

__global__ __launch_bounds__(256)
void edge_segsum(const int* __restrict__ esrc, const int* __restrict__ edst,
                 const float* __restrict__ p, const float* __restrict__ q,
                 const float* __restrict__ m, float* __restrict__ ssum,
                 int E, int NE)
{
    const int e = blockIdx.x * 8 + (threadIdx.x >> 5);
    if (e >= NE) return;
    int s, d; edge_decode(e, E, esrc, edst, s, d);
    const int c = (threadIdx.x & 31) * 4;
    const float4 pv = *reinterpret_cast<const float4*>(p + (size_t)d * D + c);
    const float4 qv = *reinterpret_cast<const float4*>(q + (size_t)s * D + c);
    const float4 mv = *reinterpret_cast<const float4*>(m + (size_t)d * D + c);
    float* sb = ssum + (size_t)d * D + c;
    atomicAdd(sb + 0, __expf(lrelu(pv.x + qv.x) - mv.x));
    atomicAdd(sb + 1, __expf(lrelu(pv.y + qv.y) - mv.y));
    atomicAdd(sb + 2, __expf(lrelu(pv.z + qv.z) - mv.z));
    atomicAdd(sb + 3, __expf(lrelu(pv.w + qv.w) - mv.w));
}

__global__ __launch_bounds__(256)
void edge_aggregate(const int* __restrict__ esrc, const int* __restrict__ edst,
                    const float* __restrict__ p, const float* __restrict__ q,
                    const float* __restrict__ m, const float* __restrict__ ssum,
                    const float* __restrict__ h, float* __restrict__ out,
                    int E, int NE)
{
    const int e = blockIdx.x * 8 + (threadIdx.x >> 5);
    if (e >= NE) return;
    int s, d; edge_decode(e, E, esrc, edst, s, d);
    const int c = (threadIdx.x & 31) * 4;
    const float4 pv = *reinterpret_cast<const float4*>(p + (size_t)d * D + c);
    const float4 qv = *reinterpret_cast<const float4*>(q + (size_t)s * D + c);
    const float4 mv = *reinterpret_cast<const float4*>(m + (size_t)d * D + c);
    const float4 sv = *reinterpret_cast<const float4*>(ssum + (size_t)d * D + c);
    const float4 hv = *reinterpret_cast<const float4*>(h + (size_t)s * D + c);
    float* ob = out + (size_t)d * D + c;
    atomicAdd(ob + 0, hv.x * (__expf(lrelu(pv.x + qv.x) - mv.x) / (sv.x + 1e-16f)));
    atomicAdd(ob + 1, hv.y * (__expf(lrelu(pv.y + qv.y) - mv.y) / (sv.y + 1e-16f)));
    atomicAdd(ob + 2, hv.z * (__expf(lrelu(pv.z + qv.z) - mv.z) / (sv.z + 1e-16f)));
    atomicAdd(ob + 3, hv.w * (__expf(lrelu(pv.w + qv.w) - mv.w) / (sv.w + 1e-16f)));
}

// ---------------------------------------------------------------------------
// Mean pool over each graph: out[b*128+d] = mean_i X[b,i,d]
// ---------------------------------------------------------------------------
__global__ __launch_bounds__(256)
void mean_pool(const float* __restrict__ X, float* __restrict__ out)
{
    const int b = blockIdx.x;
    const int t = threadIdx.x;
    const int c = t & 127;
    const int part = t >> 7;
    float acc = 0.0f;
    for (int i = part; i < NPG; i += 2)
        acc += X[((size_t)b * NPG + i) * D + c];
    __shared__ float red[256];
    red[t] = acc;
    __syncthreads();
    if (t < 128)
        out[b * D + c] = (red[t] + red[t + 128]) * (1.0f / (float)NPG);
}

// ---------------------------------------------------------------------------
// logits[n] = dot(X[n], Wp) + bp   (one wave per node)
// ---------------------------------------------------------------------------
__global__ __launch_bounds__(256)
void node_logits(const float* __restrict__ X, const float* __restrict__ Wp,
                 const float* __restrict__ bp, float* __restrict__ logits, int N)
{
    const int n = blockIdx.x * 8 + (threadIdx.x >> 5);
    if (n >= N) return;
    const int lane = threadIdx.x & 31;
    const float4 xv = *reinterpret_cast<const float4*>(X + (size_t)n * D + lane * 4);
    const float4 wv = *reinterpret_cast<const float4*>(Wp + lane * 4);
    float v = xv.x * wv.x + xv.y * wv.y + xv.z * wv.z + xv.w * wv.w;
    #pragma unroll
    for (int off = 16; off > 0; off >>= 1) v += __shfl_xor(v, off, 32);
    if (lane == 0) logits[n] = v + bp[0];
}

// ---------------------------------------------------------------------------
// u[n] = dot(tanh(SA[graph(n)] + Y[n]), vt)   (one wave per node)
// ---------------------------------------------------------------------------
__global__ __launch_bounds__(256)
void u_scores(const float* __restrict__ Y, const float* __restrict__ SA,
              const float* __restrict__ vt, float* __restrict__ u, int N)
{
    const int n = blockIdx.x * 8 + (threadIdx.x >> 5);
    if (n >= N) return;
    const int b = n >> 10;     // n / NPG
    const int lane = threadIdx.x & 31;
    const float4 yv = *reinterpret_cast<const float4*>(Y + (size_t)n * D + lane * 4);
    const float4 sv = *reinterpret_cast<const float4*>(SA + (size_t)b * D + lane * 4);
    const float4 vv = *reinterpret_cast<const float4*>(vt + lane * 4);
    float v = tanhf(yv.x + sv.x) * vv.x + tanhf(yv.y + sv.y) * vv.y
            + tanhf(yv.z + sv.z) * vv.z + tanhf(yv.w + sv.w) * vv.w;
    #pragma unroll
    for (int off = 16; off > 0; off >>= 1) v += __shfl_xor(v, off, 32);
    if (lane == 0) u[n] = v;
}

// ---------------------------------------------------------------------------
// Deterministic hash-Gumbel (stand-in for JAX threefry; deterministic per call)
// ---------------------------------------------------------------------------
__device__ __forceinline__ float gumbel_hash(unsigned a, unsigned b)
{
    unsigned x = a * 0x9E3779B9u ^ (b + 0x7F4A7C15u) * 0x85EBCA6Bu;
    x ^= x >> 16; x *= 0x7FEB352Du;
    x ^= x >> 15; x *= 0x846CA68Bu;
    x ^= x >> 16;
    const float uu = (float)(x >> 8) * (1.0f / 16777216.0f) + 1e-10f;
    return -logf(-logf(uu));
}

// ---------------------------------------------------------------------------
// Per-graph categorical sample over scores[1:NPG] (optionally excluding one
// index) + softmax probability at the sampled index.
// ---------------------------------------------------------------------------
__global__ __launch_bounds__(256)
void sample_cat(const float* __restrict__ scores, const int* __restrict__ exclude,
                unsigned salt, float* __restrict__ prob_sel, int* __restrict__ idx_out)
{
    const int b = blockIdx.x;
    const int t = threadIdx.x;
    const int excl = exclude ? exclude[b] : -1;

    float lmax = -INFINITY, pmax = -INFINITY;
    int   pidx = 1;
    for (int i = 1 + t; i < NPG; i += 256) {
        float l = scores[(size_t)b * NPG + i];
        if (i == excl) l = -INFINITY;
        lmax = fmaxf(lmax, l);
        const float g = l + gumbel_hash(salt ^ (unsigned)b, (unsigned)i);
        if (g > pmax) { pmax = g; pidx = i; }
    }

    __shared__ float sh_m[256];
    __shared__ float sh_p[256];
    __shared__ int   sh_i[256];
    sh_m[t] = lmax; sh_p[t] = pmax; sh_i[t] = pidx;
    __syncthreads();
    for (int s = 128; s > 0; s >>= 1) {
        if (t < s) {
            sh_m[t] = fmaxf(sh_m[t], sh_m[t + s]);
            if (sh_p[t + s] > sh_p[t]) { sh_p[t] = sh_p[t + s]; sh_i[t] = sh_i[t + s]; }
        }
        __syncthreads();
    }
    const float M = sh_m[0];
    const int isel = sh_i[0];
    __syncthreads();

    float se = 0.0f;
    for (int i = 1 + t; i < NPG; i += 256) {
        if (i == excl) continue;
        se += __expf(scores[(size_t)b * NPG + i] - M);
    }
    sh_m[t] = se;
    __syncthreads();
    for (int s = 128; s > 0; s >>= 1) {
        if (t < s) sh_m[t] += sh_m[t + s];
        __syncthreads();
    }
    if (t == 0) {
        const float lsel = scores[(size_t)b * NPG + isel];
        prob_sel[b] = __expf(lsel - M) / sh_m[0];
        idx_out[b]  = isel;
    }
}

// ---------------------------------------------------------------------------
// Gather sampled node embeddings: Sg[b] = X[b*NPG + idx1[b]]
// ---------------------------------------------------------------------------
__global__ void gather_sampled(const float* __restrict__ X,
                               const int* __restrict__ i1, float* __restrict__ Sg)
{
    const int b = blockIdx.x;
    const int t = threadIdx.x;   // 128 threads
    Sg[(size_t)b * D + t] = X[((size_t)b * NPG + i1[b]) * D + t];
}

// ---------------------------------------------------------------------------
// Write index1/index2/log_probs tail of the output tuple
// ---------------------------------------------------------------------------
__global__ void finalize_out(const int* __restrict__ i1, const int* __restrict__ i2,
                             const float* __restrict__ p1, const float* __restrict__ p2,
                             float* __restrict__ out)
{
    const int b = threadIdx.x;
    if (b < BATCH) {
        out[4096 + b]       = (float)i1[b];
        out[4096 + 32 + b]  = (float)i2[b];
        out[4096 + 64 + b]  = logf(p1[b] + p2[b]);
    }
}

// ---------------------------------------------------------------------------
// Host launch
// ---------------------------------------------------------------------------
extern "C" void kernel_launch(void* const* d_in, const int* in_sizes, int n_in,
                              void* d_out, int out_size, void* d_ws, size_t ws_size,
                              hipStream_t stream)
{
    const float* x    = (const float*)d_in[0];
    const int*   e_n  = (const int*)d_in[1];
    const int*   e_r0 = (const int*)d_in[2];
    const int*   e_r1 = (const int*)d_in[3];
    const float* W0l = (const float*)d_in[4];  const float* b0l = (const float*)d_in[5];
    const float* W0a = (const float*)d_in[6];  const float* b0a = (const float*)d_in[7];
    const float* W1l = (const float*)d_in[8];  const float* b1l = (const float*)d_in[9];
    const float* W1a = (const float*)d_in[10]; const float* b1a = (const float*)d_in[11];
    const float* W2l = (const float*)d_in[12]; const float* b2l = (const float*)d_in[13];
    const float* W2a = (const float*)d_in[14]; const float* b2a = (const float*)d_in[15];
    const float* Wp  = (const float*)d_in[16]; const float* bp  = (const float*)d_in[17];
    const float* Wa1 = (const float*)d_in[18];
    const float* Wa2 = (const float*)d_in[19];
    const float* vt  = (const float*)d_in[20];

    const int N  = in_sizes[0] / D;        // 32768
    const int E  = in_sizes[1] / 2;        // 262144
    const int NE = E + N;                  // with self loops
    const size_t NF = (size_t)N * D;       // feature-buffer element count

    float* ws = (float*)d_ws;
    float* X  = ws;            // running features / layer output accumulator
    float* H  = ws + 1 * NF;   // h = x @ Wl + bl
    float* P  = ws + 2 * NF;   // h @ Wa_top (dst side)
    float* Q  = ws + 3 * NF;   // h @ Wa_bot + ba (src side)
    float* MX = ws + 4 * NF;   // segment max
    float* SS = ws + 5 * NF;   // segment sum
    float* AUX = ws + 6 * NF;
    float* logits = AUX;                   // [N]
    float* u      = AUX + N;               // [N]
    float* Sg     = AUX + 2 * (size_t)N;   // [B,128]
    float* SA     = Sg + BATCH * D;        // [B,128]
    float* p1sel  = SA + BATCH * D;        // [B]
    float* p2sel  = p1sel + BATCH;         // [B]
    int*   i1     = (int*)(p2sel + BATCH); // [B]
    int*   i2     = i1 + BATCH;            // [B]

    const int gemm_blocks = (N + 127) / 128;
    const int edge_blocks = (NE + 7) / 8;
    const int init_blocks = (int)((NF + 255) / 256);
    const int node_blocks = (N + 7) / 8;

    struct LayerCfg { const int* ei; const float* Wl; const float* bl;
                      const float* Wa; const float* ba; };
    const LayerCfg layers[5] = {
        { e_n,  W0l, b0l, W0a, b0a },
        { e_r1, W1l, b1l, W1a, b1a },
        { e_r0, W2l, b2l, W2a, b2a },
        { e_r1, W1l, b1l, W1a, b1a },
        { e_r0, W2l, b2l, W2a, b2a },
    };

    for (int l = 0; l < 5; ++l) {
        const float* xin = (l == 0) ? x : X;
        const int* esrc = layers[l].ei;
        const int* edst = layers[l].ei + E;
        // h = xin @ Wl + bl
        gemm128_wmma<<<gemm_blocks, 256, 0, stream>>>(xin, layers[l].Wl, layers[l].bl, H, N);
        // p = h @ Wa[:128]   (dst contribution)
        gemm128_wmma<<<gemm_blocks, 256, 0, stream>>>(H, layers[l].Wa, nullptr, P, N);
        // q = h @ Wa[128:] + ba  (src contribution; bias folded here)
        gemm128_wmma<<<gemm_blocks, 256, 0, stream>>>(H, layers[l].Wa + 128 * D, layers[l].ba, Q, N);
        // X becomes this layer's output accumulator (xin dead after first GEMM)
        init_layer<<<init_blocks, 256, 0, stream>>>(MX, SS, X, NF);
        edge_segmax<<<edge_blocks, 256, 0, stream>>>(esrc, edst, P, Q, MX, E, NE);
        edge_segsum<<<edge_blocks, 256, 0, stream>>>(esrc, edst, P, Q, MX, SS, E, NE);
        edge_aggregate<<<edge_blocks, 256, 0, stream>>>(esrc, edst, P, Q, MX, SS, H, X, E, NE);
    }

    float* out = (float*)d_out;
    // h (mean pool) -> out[0:4096]
    mean_pool<<<BATCH, 256, 0, stream>>>(X, out);
    // logits1 and first categorical sample
    node_logits<<<node_blocks, 256, 0, stream>>>(X, Wp, bp, logits, N);
    sample_cat<<<BATCH, 256, 0, stream>>>(logits, nullptr, 0x243F6A88u, p1sel, i1);
    // attention scorer: u = tanh(sampled@Wa1 + X@Wa2) @ vt
    gather_sampled<<<BATCH, 128, 0, stream>>>(X, i1, Sg);
    gemm128_wmma<<<1, 256, 0, stream>>>(Sg, Wa1, nullptr, SA, BATCH);
    float* Y = P;  // reuse: P is dead after the layer loop
    gemm128_wmma<<<gemm_blocks, 256, 0, stream>>>(X, Wa2, nullptr, Y, N);
    u_scores<<<node_blocks, 256, 0, stream>>>(Y, SA, vt, u, N);
    // second categorical sample with index1 masked out
    sample_cat<<<BATCH, 256, 0, stream>>>(u, i1, 0x85A308D3u, p2sel, i2);
    finalize_out<<<1, 64, 0, stream>>>(i1, i2, p1sel, p2sel, out);
}